// F1Layer_85177791414420
// MI455X (gfx1250) — compile-verified
//
#include <hip/hip_runtime.h>
#include <math.h>

// F1Layer for MI455X (gfx1250): fp32 WMMA GEMMs with TDM (tensor_load_to_lds)
// double-buffered LDS staging + flash-style masked subspace attention +
// on-device 32x32 cholesky-inverse + LN epilogue + loss reductions.

#define NN 4096
#define DD 256
#define RR 32
#define KH 3
#define ETA_C 0.5f
#define COEFF_C (32.0f / (4096.0f * 0.25f))   // r / (n * eps^2)

typedef float v2f __attribute__((ext_vector_type(2)));
typedef float v8f __attribute__((ext_vector_type(8)));
typedef unsigned int v4u __attribute__((ext_vector_type(4)));
typedef int v8i __attribute__((ext_vector_type(8)));
typedef int v4i __attribute__((ext_vector_type(4)));

__device__ __forceinline__ v8f wmma_f32(v2f a, v2f b, v8f c) {
  // V_WMMA_F32_16X16X4_F32 : D = A(16x4,f32) * B(4x16,f32) + C(16x16,f32)
  return __builtin_amdgcn_wmma_f32_16x16x4_f32(false, a, false, b, (short)0, c,
                                               false, false);
}

// ---------------------------------------------------------------------------
// TDM: issue a 2-D tile load (global -> LDS) via TENSOR_LOAD_TO_LDS.
// tile_x * 4B contiguous bytes per row, tile_y rows, row stride = stride_elems
// floats. Reads past tensor_x/tensor_y return zero (replaces column guards).
// pad codes: after 2<<pad_int_code dwords insert (pad_amt_code+1) dwords of
// padding in LDS (hardware bank-conflict padding).
// ---------------------------------------------------------------------------
__device__ __forceinline__ void tdm_load_2d(unsigned lds_off, const float* gptr,
                                            unsigned tile_x, unsigned tile_y,
                                            unsigned tensor_x, unsigned tensor_y,
                                            unsigned stride_elems,
                                            unsigned pad_int_code,
                                            unsigned pad_amt_code) {
  unsigned long long ga = (unsigned long long)(size_t)gptr;
  v4u g0;
  g0[0] = 1u;  // count=1, is_restore=0, gather=0
  g0[1] = lds_off;
  g0[2] = (unsigned)(ga & 0xFFFFFFFFu);
  g0[3] = (unsigned)((ga >> 32) & 0x01FFFFFFu) | (2u << 30);  // type=2 (image)
  v8i g1;
  // data_size=2 (4B), pad_enable=1, pad_interval, pad_amount; wg_mask=0
  unsigned dw0 = (2u << 16) | (1u << 20) | (pad_int_code << 22) |
                 (pad_amt_code << 25);
  g1[0] = (int)dw0;
  g1[1] = (int)((tensor_x & 0xFFFFu) << 16);
  g1[2] = (int)((tensor_x >> 16) | ((tensor_y & 0xFFFFu) << 16));
  g1[3] = (int)((tensor_y >> 16) | (tile_x << 16));
  g1[4] = (int)(tile_y & 0xFFFFu);  // tile_dim1; tile_dim2 = 0
  g1[5] = (int)stride_elems;        // tensor_dim0_stride (low 32, elements)
  g1[6] = 0;
  g1[7] = 0;
  v4i z4 = {0, 0, 0, 0};
  v8i z8 = {0, 0, 0, 0, 0, 0, 0, 0};
  __builtin_amdgcn_tensor_load_to_lds(g0, g1, z4, z4, z8, 0);
}

// ---------------------------------------------------------------------------
// Generic fp32 WMMA GEMM: C[M,N] = alpha * A[M,K] @ B[K,N] + beta * C
// Row-major. Requires M % 64 == 0, K % 16 == 0. N arbitrary (TDM zero-fills
// OOB columns; stores guarded). Block: 256 threads = 8 waves; block tile
// 64(M) x 128(N); wave tile 16x64. LDS tiles DMA'd by wave 0 via TDM,
// double-buffered against WMMA compute.
// ---------------------------------------------------------------------------
__global__ __launch_bounds__(256) void gemm_f32(
    int M, int N, int K, const float* __restrict__ A, const float* __restrict__ B,
    float* __restrict__ C, const float* __restrict__ alphaPtr, float alphaScale,
    float beta) {
  __shared__ float As[2][64][17];   // 64 x 16 + 1 pad dword per row
  __shared__ float Bs[2][16][132];  // 16 x 128 + 4 pad dwords per row
  const int tid = threadIdx.x;
  const int lane = tid & 31;
  const int wave = tid >> 5;
  const int mT = wave & 3;   // wave row tile (16 rows each)
  const int nH = wave >> 2;  // wave col half (64 cols each)
  const int m0 = blockIdx.x * 64;
  const int n0 = blockIdx.y * 128;

  v8f acc[4] = {};

  const unsigned ldsA0 = (unsigned)(size_t)(void*)&As[0][0][0];
  const unsigned ldsA1 = (unsigned)(size_t)(void*)&As[1][0][0];
  const unsigned ldsB0 = (unsigned)(size_t)(void*)&Bs[0][0][0];
  const unsigned ldsB1 = (unsigned)(size_t)(void*)&Bs[1][0][0];

  const int nkt = K / 16;
  if (wave == 0) {
    // prologue: k-tile 0 into buffer 0
    tdm_load_2d(ldsA0, A + (size_t)m0 * K, 16, 64, (unsigned)K, 64u,
                (unsigned)K, 3u, 0u);  // pad: 16 dwords -> +1 dword
    tdm_load_2d(ldsB0, B + n0, 128, 16, (unsigned)(N - n0), 16u, (unsigned)N,
                6u, 3u);               // pad: 128 dwords -> +4 dwords
  }

  for (int kt = 0; kt < nkt; ++kt) {
    const int buf = kt & 1;
    const int k0 = kt * 16;
    if (wave == 0) {
      if (kt + 1 < nkt) {
        const int k1 = k0 + 16;
        tdm_load_2d(buf ? ldsA0 : ldsA1, A + (size_t)m0 * K + k1, 16, 64,
                    (unsigned)(K - k1), 64u, (unsigned)K, 3u, 0u);
        tdm_load_2d(buf ? ldsB0 : ldsB1, B + (size_t)k1 * N + n0, 128, 16,
                    (unsigned)(N - n0), 16u, (unsigned)N, 6u, 3u);
        __builtin_amdgcn_s_wait_tensorcnt(2);  // current tile complete
      } else {
        __builtin_amdgcn_s_wait_tensorcnt(0);
      }
    }
    __syncthreads();  // LDS tile visible to all waves
#pragma unroll
    for (int kk = 0; kk < 16; kk += 4) {
      const int ko = kk + ((lane >= 16) ? 2 : 0);
      const int ar = mT * 16 + (lane & 15);
      v2f a;
      a.x = As[buf][ar][ko];
      a.y = As[buf][ar][ko + 1];
#pragma unroll
      for (int nt = 0; nt < 4; ++nt) {
        const int bcol = nH * 64 + nt * 16 + (lane & 15);
        v2f b;
        b.x = Bs[buf][ko][bcol];
        b.y = Bs[buf][ko + 1][bcol];
        acc[nt] = wmma_f32(a, b, acc[nt]);
      }
    }
    __syncthreads();  // all waves done with this buffer before TDM reuses it
  }

  const float alpha = alphaScale * (alphaPtr ? *alphaPtr : 1.0f);
  const int crow0 = m0 + mT * 16 + ((lane >= 16) ? 8 : 0);
#pragma unroll
  for (int nt = 0; nt < 4; ++nt) {
    int col = n0 + nH * 64 + nt * 16 + (lane & 15);
    if (col < N) {
#pragma unroll
      for (int v = 0; v < 8; ++v) {
        size_t idx = (size_t)(crow0 + v) * N + col;
        float prev = (beta != 0.0f) ? beta * C[idx] : 0.0f;
        C[idx] = alpha * acc[nt][v] + prev;
      }
    }
  }
}

// ---------------------------------------------------------------------------
// Masked subspace attention (flash style): O = softmax(Q K^T + mask) @ K
// Q,K,O: [4096, 32]. mask read streaming; N x N scores never materialized.
// Block: 64 query rows, 256 threads (8 waves). S tile via WMMA into LDS,
// online softmax + rank-32 PV in VALU.
// ---------------------------------------------------------------------------
__global__ __launch_bounds__(256) void attn_kernel(
    const float* __restrict__ Q, const float* __restrict__ Km,
    const float* __restrict__ mask, float* __restrict__ O) {
  __shared__ float Qs[64][33];
  __shared__ float Ks[64][33];
  __shared__ float Ss[64][65];
  const int tid = threadIdx.x;
  const int lane = tid & 31;
  const int wave = tid >> 5;
  const int m0 = blockIdx.x * 64;

  {
    int r = tid >> 2, c = (tid & 3) * 8;
    const float* qp = Q + (size_t)(m0 + r) * RR + c;
#pragma unroll
    for (int i = 0; i < 8; ++i) Qs[r][c + i] = qp[i];
  }

  const int row = tid >> 2;  // 0..63 (softmax/PV row)
  const int q4 = tid & 3;
  const int c0 = q4 * 8;  // O column base
  float o[8];
#pragma unroll
  for (int i = 0; i < 8; ++i) o[i] = 0.0f;
  float m_run = -3.0e38f, l_run = 0.0f;
  const int mT = wave & 3, nH = wave >> 2;

  for (int jb = 0; jb < NN / 64; ++jb) {
    const int j0 = jb * 64;
    __syncthreads();  // protect Ks/Ss reuse from previous iteration
    {
      int r = tid >> 2, c = (tid & 3) * 8;
      const float* kp = Km + (size_t)(j0 + r) * RR + c;
#pragma unroll
      for (int i = 0; i < 8; ++i) Ks[r][c + i] = kp[i];
    }
    __syncthreads();
    // S = Q @ K^T : each wave computes two 16x16 tiles
#pragma unroll
    for (int t = 0; t < 2; ++t) {
      const int nt = nH * 2 + t;
      v8f s = {};
#pragma unroll
      for (int kk = 0; kk < 32; kk += 4) {
        const int ko = kk + ((lane >= 16) ? 2 : 0);
        const int ar = mT * 16 + (lane & 15);
        v2f a;
        a.x = Qs[ar][ko];
        a.y = Qs[ar][ko + 1];
        const int bcl = nt * 16 + (lane & 15);
        v2f b;  // B[k][n] = K[n][k]
        b.x = Ks[bcl][ko];
        b.y = Ks[bcl][ko + 1];
        s = wmma_f32(a, b, s);
      }
      const int lr0 = mT * 16 + ((lane >= 16) ? 8 : 0);
      const int lc = nt * 16 + (lane & 15);
      const float* mp = mask + (size_t)(m0 + lr0) * NN + j0 + lc;
#pragma unroll
      for (int v = 0; v < 8; ++v) {
        float mk = mp[(size_t)v * NN];
        Ss[lr0 + v][lc] = s[v] + (1.0f - mk) * (-1.0e9f);
      }
    }
    __syncthreads();
    // online softmax over this 64-column block (4 threads per row)
    {
      const int jbase = q4 * 16;
      float mloc = -3.0e38f;
#pragma unroll
      for (int j = 0; j < 16; ++j) mloc = fmaxf(mloc, Ss[row][jbase + j]);
      mloc = fmaxf(mloc, __shfl_xor(mloc, 1, 32));
      mloc = fmaxf(mloc, __shfl_xor(mloc, 2, 32));
      float m_new = fmaxf(m_run, mloc);
      float scale = __expf(m_run - m_new);
      float ladd = 0.0f;
#pragma unroll
      for (int j = 0; j < 16; ++j) {
        float p = __expf(Ss[row][jbase + j] - m_new);
        Ss[row][jbase + j] = p;
        ladd += p;
      }
      ladd += __shfl_xor(ladd, 1, 32);
      ladd += __shfl_xor(ladd, 2, 32);
      l_run = l_run * scale + ladd;
      m_run = m_new;
#pragma unroll
      for (int i = 0; i < 8; ++i) o[i] *= scale;
    }
    __syncthreads();
    // O += P @ K  (head dim 32; each thread owns 8 columns of its row)
    for (int j = 0; j < 64; ++j) {
      float p = Ss[row][j];
#pragma unroll
      for (int i = 0; i < 8; ++i) o[i] += p * Ks[j][c0 + i];
    }
  }
  float inv = 1.0f / l_run;
  float* op = O + (size_t)(m0 + row) * RR + c0;
#pragma unroll
  for (int i = 0; i < 8; ++i) op[i] = o[i] * inv;
}

// ---------------------------------------------------------------------------
// M = I + coeff * (Q^T Q),  Q:[4096,32]  ->  Mout[32*32]
// ---------------------------------------------------------------------------
__global__ __launch_bounds__(1024) void gram_kernel(const float* __restrict__ Q,
                                                    float* __restrict__ Mout) {
  __shared__ float ch[32][33];
  const int tid = threadIdx.x;
  const int i = tid >> 5, j = tid & 31;
  float acc = 0.0f;
  for (int c = 0; c < NN / 32; ++c) {
    __syncthreads();
    ch[i][j] = Q[(size_t)(c * 32 + i) * RR + j];
    __syncthreads();
#pragma unroll
    for (int t = 0; t < 32; ++t) acc += ch[t][i] * ch[t][j];
  }
  Mout[i * 32 + j] = ((i == j) ? 1.0f : 0.0f) + COEFF_C * acc;
}

// ---------------------------------------------------------------------------
// 32x32 SPD cholesky inverse, one wave. Minv = (L L^T)^-1 = L^-T L^-1.
// ---------------------------------------------------------------------------
__global__ __launch_bounds__(32) void cholinv_kernel(
    const float* __restrict__ Min, float* __restrict__ Minv) {
  __shared__ float Ms[32][33];
  __shared__ float Wl[32][33];
  const int t = threadIdx.x;
  for (int j = 0; j < 32; ++j) Ms[t][j] = Min[t * 32 + j];
  __syncthreads();
  for (int k = 0; k < 32; ++k) {
    if (t == k) Ms[k][k] = sqrtf(Ms[k][k]);
    __syncthreads();
    if (t > k) Ms[t][k] /= Ms[k][k];
    __syncthreads();
    for (int j = k + 1; j < 32; ++j)
      if (t >= j) Ms[t][j] -= Ms[t][k] * Ms[j][k];
    __syncthreads();
  }
  // column t of W = L^-1 (forward substitution, thread-local)
  float wc[32];
  for (int i = 0; i < 32; ++i) {
    if (i < t) {
      wc[i] = 0.0f;
    } else {
      float s = (i == t) ? 1.0f : 0.0f;
      for (int p = t; p < i; ++p) s -= Ms[i][p] * wc[p];
      wc[i] = s / Ms[i][i];
    }
  }
  for (int i = 0; i < 32; ++i) Wl[i][t] = wc[i];
  __syncthreads();
  for (int j = 0; j < 32; ++j) {
    float s = 0.0f;
    for (int kk = 0; kk < 32; ++kk) s += Wl[kk][t] * Wl[kk][j];
    Minv[t * 32 + j] = s;
  }
}

// w = softmax(hop_weights), laps = softplus(lambda); also writes w to output.
__global__ void scal_kernel(const float* __restrict__ hw,
                            const float* __restrict__ ll, float* __restrict__ scal,
                            float* __restrict__ outW) {
  if (threadIdx.x == 0) {
    float m = fmaxf(hw[0], fmaxf(hw[1], hw[2]));
    float e0 = expf(hw[0] - m), e1 = expf(hw[1] - m), e2 = expf(hw[2] - m);
    float s = e0 + e1 + e2;
    scal[0] = e0 / s;
    scal[1] = e1 / s;
    scal[2] = e2 / s;
    for (int k = 0; k < 3; ++k) {
      float x = ll[k];
      scal[3 + k] = (x > 20.0f) ? x : log1pf(expf(x));
    }
    outW[0] = scal[0];
    outW[1] = scal[1];
    outW[2] = scal[2];
  }
}

// Ukt[k][r][d] = U[k][d][r]
__global__ void transu_kernel(const float* __restrict__ U,
                              float* __restrict__ Ukt) {
  int idx = blockIdx.x * blockDim.x + threadIdx.x;
  if (idx < KH * RR * DD) {
    int k = idx / (RR * DD);
    int rem = idx % (RR * DD);
    int r = rem / DD, d = rem % DD;
    Ukt[idx] = U[(size_t)k * DD * RR + (size_t)d * RR + r];
  }
}

// Hmix = laps0*H + laps1*H1 + laps2*H2
__global__ void hmix_kernel(const float* __restrict__ H,
                            const float* __restrict__ H1,
                            const float* __restrict__ H2,
                            const float* __restrict__ scal,
                            float* __restrict__ Hmix) {
  int idx = blockIdx.x * blockDim.x + threadIdx.x;
  if (idx < NN * DD)
    Hmix[idx] = scal[3] * H[idx] + scal[4] * H1[idx] + scal[5] * H2[idx];
}

// soft-threshold + residual + LayerNorm; one block per row.
__global__ __launch_bounds__(256) void epilogue_kernel(
    const float* __restrict__ H, const float* __restrict__ AGG,
    const float* __restrict__ thr, const float* __restrict__ gamma,
    const float* __restrict__ beta, float* __restrict__ Hout) {
  __shared__ float s1[256], s2[256];
  const int row = blockIdx.x, d = threadIdx.x;
  const size_t idx = (size_t)row * DD + d;
  float h = H[idx];
  float hh = h + AGG[idx];  // AGG already holds ETA*(agg - lap_total)
  float ab = fabsf(hh) - thr[d];
  float st = (ab > 0.0f) ? copysignf(ab, hh) : 0.0f;
  float out = st + h;
  s1[d] = out;
  s2[d] = out * out;
  __syncthreads();
  for (int off = 128; off > 0; off >>= 1) {
    if (d < off) {
      s1[d] += s1[d + off];
      s2[d] += s2[d + off];
    }
    __syncthreads();
  }
  float mu = s1[0] * (1.0f / DD);
  float var = s2[0] * (1.0f / DD) - mu * mu;
  Hout[idx] = (out - mu) * rsqrtf(var + 1e-5f) * gamma[d] + beta[d];
}

// partial dot products for lap_smooth = sum(H_out * (L @ H_out))
__global__ __launch_bounds__(256) void dotpart_kernel(
    const float* __restrict__ X, const float* __restrict__ Y,
    float* __restrict__ part) {
  __shared__ float sm[256];
  const int t = threadIdx.x;
  const size_t stride = (size_t)gridDim.x * 256;
  float acc = 0.0f;
  for (size_t i = (size_t)blockIdx.x * 256 + t; i < (size_t)NN * DD; i += stride)
    acc += X[i] * Y[i];
  sm[t] = acc;
  __syncthreads();
  for (int off = 128; off > 0; off >>= 1) {
    if (t < off) sm[t] += sm[t + off];
    __syncthreads();
  }
  if (t == 0) part[blockIdx.x] = sm[0];
}

// orthogonality loss: one block per (k,l) pair, G = Uk^T Ul (32x32)
__global__ __launch_bounds__(1024) void orth_kernel(const float* __restrict__ U,
                                                    float* __restrict__ part) {
  __shared__ float Ak[32][33];
  __shared__ float Al[32][33];
  __shared__ float red[1024];
  const int p = blockIdx.x;
  int k, l;
  if (p == 0) { k = 0; l = 0; }
  else if (p == 1) { k = 0; l = 1; }
  else if (p == 2) { k = 0; l = 2; }
  else if (p == 3) { k = 1; l = 1; }
  else if (p == 4) { k = 1; l = 2; }
  else { k = 2; l = 2; }
  const int tid = threadIdx.x;
  const int i = tid >> 5, j = tid & 31;
  float acc = 0.0f;
  for (int c = 0; c < DD / 32; ++c) {
    __syncthreads();
    Ak[i][j] = U[(size_t)k * DD * RR + (size_t)(c * 32 + i) * RR + j];
    Al[i][j] = U[(size_t)l * DD * RR + (size_t)(c * 32 + i) * RR + j];
    __syncthreads();
#pragma unroll
    for (int t = 0; t < 32; ++t) acc += Ak[t][i] * Al[t][j];
  }
  float g = acc - ((k == l && i == j) ? 1.0f : 0.0f);
  red[tid] = g * g;
  __syncthreads();
  for (int off = 512; off > 0; off >>= 1) {
    if (tid < off) red[tid] += red[tid + off];
    __syncthreads();
  }
  if (tid == 0) part[p] = red[0];
}

__global__ void finalize_kernel(const float* __restrict__ lpart,
                                const float* __restrict__ opart,
                                float* __restrict__ outOrth,
                                float* __restrict__ outLap) {
  if (threadIdx.x == 0) {
    float s = 0.0f;
    for (int i = 0; i < 6; ++i) s += opart[i];
    *outOrth = s;
    float l = 0.0f;
    for (int i = 0; i < 1024; ++i) l += lpart[i];
    *outLap = l;
  }
}

extern "C" void kernel_launch(void* const* d_in, const int* in_sizes, int n_in,
                              void* d_out, int out_size, void* d_ws,
                              size_t ws_size, hipStream_t stream) {
  (void)in_sizes; (void)n_in; (void)out_size; (void)ws_size;
  const float* H = (const float*)d_in[0];
  const float* A = (const float*)d_in[1];
  const float* msk = (const float*)d_in[2];
  const float* L = (const float*)d_in[3];
  const float* U = (const float*)d_in[4];
  const float* lam = (const float*)d_in[5];
  const float* hw = (const float*)d_in[6];
  const float* thr = (const float*)d_in[7];
  const float* gm = (const float*)d_in[8];
  const float* bt = (const float*)d_in[9];

  float* out = (float*)d_out;
  const size_t ND = (size_t)NN * DD, NR = (size_t)NN * RR;
  float* Hout = out;            // [4096,256]
  float* outOrth = out + ND;    // scalar
  float* outLap = outOrth + 1;  // scalar
  float* outW = outLap + 1;     // [3]

  float* ws = (float*)d_ws;
  float* H1 = ws;
  float* H2 = H1 + ND;
  float* AGG = H2 + ND;
  float* HMIX = AGG + ND;
  float* LHO = HMIX + ND;
  float* Qb = LHO + ND;
  float* Kb = Qb + NR;
  float* Ob = Kb + NR;
  float* UKT = Ob + NR;
  float* GM = UKT + (size_t)KH * RR * DD;
  float* MINV = GM + RR * RR;
  float* SCAL = MINV + RR * RR;
  float* LPART = SCAL + 8;
  float* OPART = LPART + 1024;

  const dim3 blk(256);
  const dim3 gBig(NN / 64, (DD + 127) / 128);  // 64 x 2
  const dim3 gSkinny(NN / 64, 1);

  scal_kernel<<<1, 32, 0, stream>>>(hw, lam, SCAL, outW);
  transu_kernel<<<(KH * RR * DD + 255) / 256, blk, 0, stream>>>(U, UKT);

  // multi-hop propagation
  gemm_f32<<<gBig, blk, 0, stream>>>(NN, DD, NN, A, H, H1, nullptr, 1.0f, 0.0f);
  gemm_f32<<<gBig, blk, 0, stream>>>(NN, DD, NN, A, H1, H2, nullptr, 1.0f, 0.0f);

  for (int k = 0; k < KH; ++k) {
    const float* Hk = (k == 0) ? H : ((k == 1) ? H1 : H2);
    // Q = Hk @ Uk
    gemm_f32<<<gSkinny, blk, 0, stream>>>(NN, RR, DD, Hk,
                                          U + (size_t)k * DD * RR, Qb, nullptr,
                                          1.0f, 0.0f);
    gram_kernel<<<1, 1024, 0, stream>>>(Qb, GM);
    cholinv_kernel<<<1, 32, 0, stream>>>(GM, MINV);
    // K = Q @ Minv
    gemm_f32<<<gSkinny, blk, 0, stream>>>(NN, RR, RR, Qb, MINV, Kb, nullptr,
                                          1.0f, 0.0f);
    attn_kernel<<<NN / 64, blk, 0, stream>>>(Qb, Kb, msk, Ob);
    // AGG += ETA * w_k * (O @ Uk^T)
    gemm_f32<<<gBig, blk, 0, stream>>>(NN, DD, RR, Ob, UKT + (size_t)k * RR * DD,
                                       AGG, SCAL + k, ETA_C,
                                       (k == 0) ? 0.0f : 1.0f);
  }

  // AGG -= ETA * L @ (sum_k laps_k * Hk)
  hmix_kernel<<<(int)((ND + 255) / 256), blk, 0, stream>>>(H, H1, H2, SCAL, HMIX);
  gemm_f32<<<gBig, blk, 0, stream>>>(NN, DD, NN, L, HMIX, AGG, nullptr, -ETA_C,
                                     1.0f);

  epilogue_kernel<<<NN, 256, 0, stream>>>(H, AGG, thr, gm, bt, Hout);

  // lap_smooth = sum(H_out * (L @ H_out))
  gemm_f32<<<gBig, blk, 0, stream>>>(NN, DD, NN, L, Hout, LHO, nullptr, 1.0f,
                                     0.0f);
  dotpart_kernel<<<1024, blk, 0, stream>>>(Hout, LHO, LPART);
  orth_kernel<<<6, 1024, 0, stream>>>(U, OPART);
  finalize_kernel<<<1, 32, 0, stream>>>(LPART, OPART, outOrth, outLap);
}